// ConvolutionalCapsule_58540404244860
// MI455X (gfx1250) — compile-verified
//
#include <hip/hip_runtime.h>
#include <math.h>

// ---------------------------------------------------------------------------
// ConvolutionalCapsule (EM routing) for MI455X / gfx1250, wave32.
//
// Vote transform = per-child GEMM with K=4  ->  V_WMMA_F32_16X16X4_F32.
// Routing = bandwidth-bound streaming over the 236MB votes tensor in d_ws;
// rr posteriors live entirely in LDS (36KB << 320KB/WGP), so each routing
// iteration costs exactly one (M-step) or two (M+E) coalesced passes.
// ---------------------------------------------------------------------------

typedef float v2f __attribute__((ext_vector_type(2)));
typedef float v8f __attribute__((ext_vector_type(8)));

#define EPSF      1e-9f
#define NB        4      // batch
#define HW        12     // input H == W
#define NI        288    // 9 * 32 children per position
#define NO        32     // parent capsules
#define NP        16     // pose components (4x4)
#define SS        10     // output spatial size
#define NPOS      400    // NB * SS * SS
#define POSE_OUT_ELEMS (NPOS * NO * NP)   // 204800

// ===========================================================================
// Kernel 1: votes[p][i][o][a*4+c] = sum_b pose_tile[p][i][a][b] * w[i][o][b][c]
// One wave = one 16x16 tile:  M rows = 4 positions x 4 pose-rows,
//                             N cols = 4 parents  x 4 pose-cols,  K = 4.
// blockIdx.x = i*100 + ptile ;  wave-in-block = parent tile (otile 0..7).
// ===========================================================================
__global__ __launch_bounds__(256)
void votes_wmma_kernel(const float* __restrict__ pose,   // [4,12,12,32,16]
                       const float* __restrict__ w,      // [288,32,4,4]
                       float* __restrict__ votes)        // [400,288,32,16]
{
    const int blk   = blockIdx.x;
    const int i     = blk / 100;          // child index 0..287
    const int ptile = blk % 100;          // 4 positions per tile
    const int lane  = threadIdx.x & 31;
    const int otile = threadIdx.x >> 5;   // 0..7 -> parents otile*4 .. +3

    const int t  = i >> 5;                // 3x3 tap index
    const int ci = i & 31;                // child capsule within tap
    const int ti = t / 3, tj = t % 3;

    // ---- A operand: 16x4 (M x K) f32, ISA wave32 layout --------------------
    // lane&15 = M row, lane>>4 selects K half {0,1}; VGPR0=K0/K2, VGPR1=K1/K3.
    const int m     = lane & 15;
    const int khalf = lane >> 4;
    const int k0    = khalf * 2;
    {
        // position/row for this lane's A element
    }
    const int pa   = ptile * 4 + (m >> 2);
    const int arow = m & 3;
    const int apb  = pa / 100;
    const int aprm = pa % 100;
    const int apy  = aprm / 10, apx = aprm % 10;
    const float* ap = pose +
        (((((size_t)apb * HW + (apy + ti)) * HW + (apx + tj)) * 32 + ci) * 16
         + arow * 4 + k0);
    v2f A; A.x = ap[0]; A.y = ap[1];          // contiguous b64 load

    // ---- B operand: 4x16 (K x N) f32 --------------------------------------
    // lane&15 = N col, same K half split; row of B striped across 16 lanes.
    const int n  = lane & 15;
    const int oo = otile * 4 + (n >> 2);
    const int cc = n & 3;
    const float* bp = w + ((((size_t)i * NO + oo) * 4 + k0) * 4 + cc);
    v2f Bv; Bv.x = bp[0]; Bv.y = bp[4];       // k0 and k0+1 (stride 4 floats)

    // ---- D = A x B (C = 0) -------------------------------------------------
    v8f C = {};
    C = __builtin_amdgcn_wmma_f32_16x16x4_f32(
            /*neg_a=*/false, A, /*neg_b=*/false, Bv,
            /*c_mod=*/(short)0, C, /*reuse_a=*/false, /*reuse_b=*/false);

    // ---- scatter D: vgpr j holds M = j + khalf*8, N = lane&15 --------------
    #pragma unroll
    for (int j = 0; j < 8; ++j) {
        const int dm = j + khalf * 8;
        const int dp = ptile * 4 + (dm >> 2);
        const int da = dm & 3;
        votes[((((size_t)dp * NI + i) * NO + oo) * NP) + da * 4 + cc] = C[j];
    }
}

// ===========================================================================
// Kernel 2: EM routing, one block (256 threads) per spatial position.
// rr kept in LDS; votes streamed coalesced from d_ws.
// ===========================================================================
__global__ __launch_bounds__(256)
void em_routing_kernel(const float* __restrict__ votes,      // [400,288,32,16]
                       const float* __restrict__ activation, // [4,12,12,32]
                       const float* __restrict__ beta_v,     // [32]
                       const float* __restrict__ beta_a,     // [32]
                       float* __restrict__ out)              // pose|act concat
{
    __shared__ float s_ia[NI];            // child activations (tiled gather)
    __shared__ float s_rr[NI * NO];       // routing posteriors  (36 KB)
    __shared__ float s_mean[NO * NP];
    __shared__ float s_std [NO * NP];
    __shared__ float s_i2v [NO * NP];     // 1 / (2*std^2)
    __shared__ float s_scr [NO * NP];     // cost_h scratch
    __shared__ float s_cost[NO];
    __shared__ float s_act [NO];
    __shared__ float s_logs[NO];          // sum_pc log(std+eps)

    const int p   = blockIdx.x;
    const int tid = threadIdx.x;
    const int pb  = p / 100;
    const int prm = p % 100;
    const int py  = prm / 10, px = prm % 10;

    // gather child activations for this position's 3x3 receptive field
    for (int i = tid; i < NI; i += 256) {
        const int t = i >> 5, ci = i & 31;
        const int ti = t / 3, tj = t % 3;
        s_ia[i] = activation[(((size_t)pb * HW + (py + ti)) * HW + (px + tj)) * 32 + ci];
    }
    __syncthreads();

    const float* vp = votes + (size_t)p * NI * (NO * NP);

    // each thread owns two (o, pose-component) slots
    const int slot0 = tid;
    const int slot1 = tid + 256;
    const int o0 = slot0 >> 4;
    const int o1 = slot1 >> 4;

    // inv_temp = 0.01 * (1 - 0.95^(it+1))
    const float inv_temps[3] = {0.0005f, 0.000975f, 0.00142625f};

    for (int it = 0; it < 3; ++it) {
        // ---------------- M-step: single coalesced pass over votes ---------
        float rs0 = 0.f, rv0 = 0.f, rq0 = 0.f;
        float rs1 = 0.f, rv1 = 0.f, rq1 = 0.f;
        for (int i = 0; i < NI; ++i) {
            const float iai = s_ia[i];
            float r0, r1;
            if (it == 0) { r0 = 1.0f / NO; r1 = 1.0f / NO; }
            else         { r0 = s_rr[i * NO + o0]; r1 = s_rr[i * NO + o1]; }
            const float w0 = r0 * iai;
            const float w1 = r1 * iai;
            const float v0 = vp[(size_t)i * (NO * NP) + slot0];
            const float v1 = vp[(size_t)i * (NO * NP) + slot1];
            rs0 += w0; rv0 = fmaf(w0, v0, rv0); rq0 = fmaf(w0 * v0, v0, rq0);
            rs1 += w1; rv1 = fmaf(w1, v1, rv1); rq1 = fmaf(w1 * v1, v1, rq1);
        }
        const float m0 = rv0 / rs0, m1 = rv1 / rs1;
        const float var0 = fmaxf(rq0 / rs0 - m0 * m0, 0.0f);
        const float var1 = fmaxf(rq1 / rs1 - m1 * m1, 0.0f);
        const float st0 = sqrtf(var0), st1 = sqrtf(var1);
        s_mean[slot0] = m0;  s_mean[slot1] = m1;
        s_std [slot0] = st0; s_std [slot1] = st1;
        s_i2v [slot0] = 1.0f / (2.0f * var0 + 1e-30f);
        s_i2v [slot1] = 1.0f / (2.0f * var1 + 1e-30f);
        s_scr [slot0] = (beta_v[o0] + logf(st0 + EPSF)) * rs0;
        s_scr [slot1] = (beta_v[o1] + logf(st1 + EPSF)) * rs1;
        __syncthreads();

        if (tid < NO) {
            float cs = 0.f, ls = 0.f;
            for (int pc = 0; pc < NP; ++pc) {
                cs += s_scr[tid * NP + pc];
                ls += logf(s_std[tid * NP + pc] + EPSF);
            }
            s_cost[tid] = cs;
            s_logs[tid] = ls;
        }
        __syncthreads();
        if (tid < NO) {
            float cm = 0.f;
            for (int o = 0; o < NO; ++o) cm += s_cost[o];
            cm *= (1.0f / NO);
            float cv = 0.f;
            for (int o = 0; o < NO; ++o) { const float d = s_cost[o] - cm; cv += d * d; }
            const float cstd = sqrtf(cv * (1.0f / NO));
            const float x = inv_temps[it] *
                            (beta_a[tid] + (cm - s_cost[tid]) / (cstd + EPSF));
            s_act[tid] = 1.0f / (1.0f + expf(-x));
        }
        __syncthreads();

        // ---------------- E-step (skipped on final iteration) --------------
        if (it < 2) {
            const int o  = tid & 31;   // lane == parent capsule
            const int wv = tid >> 5;   // wave-in-block -> child stride
            const float la = logf(s_act[o] + EPSF);
            for (int ibase = 0; ibase < NI; ibase += 8) {
                const int i = ibase + wv;
                float op = 0.f;
                #pragma unroll
                for (int pc = 0; pc < NP; ++pc) {
                    const float v = vp[(size_t)i * (NO * NP) + o * NP + pc];
                    const float d = v - s_mean[o * NP + pc];
                    op = fmaf(d * d, s_i2v[o * NP + pc], op);
                }
                float zz = la - op - s_logs[o];
                // softmax over o across the 32 lanes of this wave (wave32)
                float mx = zz;
                for (int off = 16; off > 0; off >>= 1)
                    mx = fmaxf(mx, __shfl_xor(mx, off, 32));
                const float e = expf(zz - mx);
                float se = e;
                for (int off = 16; off > 0; off >>= 1)
                    se += __shfl_xor(se, off, 32);
                s_rr[i * NO + o] = e / se;
            }
            __syncthreads();
        }
    }

    // ---------------- outputs: pose [400,32,16] then act [400,32] ----------
    out[(size_t)p * (NO * NP) + slot0] = s_mean[slot0];
    out[(size_t)p * (NO * NP) + slot1] = s_mean[slot1];
    if (tid < NO) out[POSE_OUT_ELEMS + (size_t)p * NO + tid] = s_act[tid];
}

// ===========================================================================
extern "C" void kernel_launch(void* const* d_in, const int* in_sizes, int n_in,
                              void* d_out, int out_size, void* d_ws, size_t ws_size,
                              hipStream_t stream) {
    const float* pose       = (const float*)d_in[0]; // [4,12,12,32,4,4]
    const float* activation = (const float*)d_in[1]; // [4,12,12,32]
    const float* w          = (const float*)d_in[2]; // [1,288,32,4,4]
    const float* beta_v     = (const float*)d_in[3]; // [1,1,1,32]
    const float* beta_a     = (const float*)d_in[4]; // [1,1,1,32]
    float* out   = (float*)d_out;                    // 204800 + 12800 floats
    float* votes = (float*)d_ws;                     // [400,288,32,16] f32

    // Kernel 1: 288 children x 100 position-tiles; 8 waves/block = 8 parent tiles
    votes_wmma_kernel<<<dim3(NI * 100), dim3(256), 0, stream>>>(pose, w, votes);
    // Kernel 2: one block per spatial position
    em_routing_kernel<<<dim3(NPOS), dim3(256), 0, stream>>>(
        votes, activation, beta_v, beta_a, out);
    (void)in_sizes; (void)n_in; (void)out_size; (void)ws_size;
}